// MyHGTLayerSlice_60241211293941
// MI455X (gfx1250) — compile-verified
//
#include <hip/hip_runtime.h>
#include <cstdint>
#include <cstddef>

// ---------------------------------------------------------------------------
// HGT layer slice for MI455X (gfx1250), f32 WMMA (V_WMMA_F32_16X16X4_F32).
// Nodes bucketed by type, edges bucketed by relation so each 16-row WMMA
// tile uses a single weight matrix. All scratch lives in d_ws.
// ---------------------------------------------------------------------------

constexpr int N = 50000, E = 500000, IN = 128, OUT = 128, T = 8, R = 8;
constexpr float INV_SQRT_DK = 0.08838834764831845f; // 1/sqrt(128)

constexpr int NODE_SLOTS = N + 16 * T;        // padded bucket slots (50128)
constexpr int NODE_TILES = NODE_SLOTS / 16;   // 3133
constexpr int EDGE_SLOTS = E + 16 * R;        // 500128
constexpr int EDGE_TILES = EDGE_SLOTS / 16;   // 31258

typedef float v2f __attribute__((ext_vector_type(2)));
typedef float v8f __attribute__((ext_vector_type(8)));

// ---- workspace layout ------------------------------------------------------
constexpr size_t wal(size_t x) { return (x + 255) & ~size_t(255); }
constexpr size_t NBYTES    = size_t(N) * OUT * sizeof(float);
constexpr size_t OFF_K     = 0;
constexpr size_t OFF_Q     = wal(OFF_K + NBYTES);
constexpr size_t OFF_V     = wal(OFF_Q + NBYTES);
constexpr size_t OFF_AGG   = wal(OFF_V + NBYTES);
constexpr size_t OFF_SC    = wal(OFF_AGG + NBYTES);                 // E floats (score -> exp)
constexpr size_t OFF_MU    = wal(OFF_SC + size_t(E) * 4);           // N*R uint (ordered max)
constexpr size_t OFF_DEN   = wal(OFF_MU + size_t(N) * R * 4);       // N*R floats
constexpr size_t OFF_CNT   = wal(OFF_DEN + size_t(N) * R * 4);      // 32 ints: ncnt8,ncur8,ecnt8,ecur8
constexpr size_t OFF_NPERM = wal(OFF_CNT + 32 * 4);
constexpr size_t OFF_TTYPE = wal(OFF_NPERM + size_t(NODE_SLOTS) * 4);
constexpr size_t OFF_EPERM = wal(OFF_TTYPE + size_t(NODE_TILES) * 4);
constexpr size_t OFF_TREL  = wal(OFF_EPERM + size_t(EDGE_SLOTS) * 4);
constexpr size_t WS_NEEDED = wal(OFF_TREL + size_t(EDGE_TILES) * 4);

// ---- ordered-float encoding for atomicMax over signed floats ---------------
__device__ __forceinline__ unsigned fkey(float f) {
  unsigned b = __float_as_uint(f);
  return (b & 0x80000000u) ? ~b : (b | 0x80000000u);
}
__device__ __forceinline__ float fdec(unsigned u) {
  unsigned b = (u & 0x80000000u) ? (u & 0x7FFFFFFFu) : ~u;
  return __uint_as_float(b);
}

// ---- 16x16 f32 WMMA tile: D = sh[16x128] @ W[:, ctile*16 : +16]  -----------
// A layout (ISA 7.12.2, 32-bit 16x4): lane&15 = M row; lanes>=16 offset K by 2
// v0 = K=koff, v1 = K=koff+1, koff = (lane>>4)*2. B mirrors with lane&15 = N.
__device__ __forceinline__ v8f wmma_tile_f32(const float* __restrict__ W,
                                             const float (&sh)[16][132],
                                             int lane, int ctile) {
  const int row  = lane & 15;
  const int koff = (lane >> 4) << 1;
  const float* Wc = W + ctile * 16 + row;   // column (N) index = ctile*16+row
  v8f acc = {0.f, 0.f, 0.f, 0.f, 0.f, 0.f, 0.f, 0.f};
#pragma unroll 4
  for (int kb = 0; kb < IN; kb += 4) {
    v2f a, b;
    a.x = sh[row][kb + koff];
    a.y = sh[row][kb + koff + 1];
    b.x = Wc[(size_t)(kb + koff) * OUT];
    b.y = Wc[(size_t)(kb + koff + 1) * OUT];
    acc = __builtin_amdgcn_wmma_f32_16x16x4_f32(false, a, false, b,
                                                (short)0, acc, false, false);
  }
  return acc;
}

// ---------------------------------------------------------------------------
__global__ void hgt_init(float* agg, unsigned* mu, float* den, int* cnt,
                         int* nperm, int* eperm, int* ttype, int* trel) {
  const int stride = gridDim.x * blockDim.x;
  const int i0 = blockIdx.x * blockDim.x + threadIdx.x;
  for (int i = i0; i < N * OUT; i += stride) agg[i] = 0.f;
  for (int i = i0; i < N * R; i += stride) { mu[i] = 0u; den[i] = 0.f; }
  for (int i = i0; i < 32; i += stride) cnt[i] = 0;
  for (int i = i0; i < NODE_SLOTS; i += stride) nperm[i] = -1;
  for (int i = i0; i < EDGE_SLOTS; i += stride) eperm[i] = -1;
  for (int i = i0; i < NODE_TILES; i += stride) ttype[i] = -1;
  for (int i = i0; i < EDGE_TILES; i += stride) trel[i] = -1;
}

__global__ void hgt_count(const int* __restrict__ node_type,
                          const int* __restrict__ edge_type, int* cnt) {
  const int i = blockIdx.x * blockDim.x + threadIdx.x;
  if (i < N) atomicAdd(&cnt[node_type[i]], 1);
  if (i < E) atomicAdd(&cnt[16 + edge_type[i]], 1);
}

__global__ void hgt_scan(int* cnt, int* ttype, int* trel) {
  __shared__ int noff[T + 1];
  __shared__ int eoff[R + 1];
  if (threadIdx.x == 0) {
    int acc = 0;
    for (int t = 0; t < T; ++t) {
      noff[t] = acc; cnt[8 + t] = acc;           // scatter cursor = offset
      acc += (cnt[t] + 15) & ~15;                // pad bucket to 16
    }
    noff[T] = acc;
    acc = 0;
    for (int r = 0; r < R; ++r) {
      eoff[r] = acc; cnt[24 + r] = acc;
      acc += (cnt[16 + r] + 15) & ~15;
    }
    eoff[R] = acc;
  }
  __syncthreads();
  for (int i = threadIdx.x; i < NODE_TILES; i += blockDim.x) {
    int t = -1;
    for (int tt = 0; tt < T; ++tt)
      if (i * 16 >= noff[tt] && i * 16 < noff[tt + 1]) t = tt;
    ttype[i] = t;
  }
  for (int i = threadIdx.x; i < EDGE_TILES; i += blockDim.x) {
    int r = -1;
    for (int rr = 0; rr < R; ++rr)
      if (i * 16 >= eoff[rr] && i * 16 < eoff[rr + 1]) r = rr;
    trel[i] = r;
  }
}

__global__ void hgt_scatter(const int* __restrict__ node_type,
                            const int* __restrict__ edge_type,
                            int* cnt, int* nperm, int* eperm) {
  const int i = blockIdx.x * blockDim.x + threadIdx.x;
  if (i < N) { int p = atomicAdd(&cnt[8 + node_type[i]], 1);  nperm[p] = i; }
  if (i < E) { int p = atomicAdd(&cnt[24 + edge_type[i]], 1); eperm[p] = i; }
}

// ---- K/Q/V projections: per-type 16-node tiles -----------------------------
__global__ __launch_bounds__(32)
void hgt_kqv(const float* __restrict__ h, const int* __restrict__ nperm,
             const int* __restrict__ ttype,
             const float* __restrict__ kW, const float* __restrict__ qW,
             const float* __restrict__ vW,
             float* __restrict__ ko, float* __restrict__ qo,
             float* __restrict__ vo) {
  __shared__ float sh[16][132];
  __shared__ int nd[16];
  const int t = ttype[blockIdx.x];
  if (t < 0) return;
  const int lane = threadIdx.x;
  if (lane < 16) nd[lane] = nperm[blockIdx.x * 16 + lane];
  __syncthreads();
#pragma unroll
  for (int m = 0; m < 16; ++m) {
    const int n = nd[m];
    float4 val = make_float4(0.f, 0.f, 0.f, 0.f);
    if (n >= 0) val = ((const float4*)h)[(size_t)n * 32 + lane];
    *(float4*)&sh[m][lane * 4] = val;
  }
  __syncthreads();
  const float* Ws[3] = {kW + (size_t)t * IN * OUT, qW + (size_t)t * IN * OUT,
                        vW + (size_t)t * IN * OUT};
  float* Os[3] = {ko, qo, vo};
  __builtin_prefetch(Ws[0], 0, 3);
  __builtin_prefetch(Ws[1], 0, 3);
  __builtin_prefetch(Ws[2], 0, 3);
  const int row = lane & 15, hi = lane >> 4;
  for (int w = 0; w < 3; ++w) {
    for (int c = 0; c < 8; ++c) {
      v8f acc = wmma_tile_f32(Ws[w], sh, lane, c);
#pragma unroll
      for (int i = 0; i < 8; ++i) {
        const int m = i + (hi << 3);           // D: lanes>=16 hold rows M+8
        const int n = nd[m];
        if (n >= 0) Os[w][(size_t)n * OUT + c * 16 + row] = acc[i];
      }
    }
  }
}

// ---- attention scores: per-relation 16-edge tiles --------------------------
__global__ __launch_bounds__(32)
void hgt_score(const float* __restrict__ kb, const float* __restrict__ qb,
               const int* __restrict__ adj, const int* __restrict__ eperm,
               const int* __restrict__ trel, const float* __restrict__ ratt,
               const float* __restrict__ pri, float* __restrict__ sc,
               unsigned* __restrict__ mu) {
  __shared__ float shk[16][132];
  __shared__ float shq[16][132];
  __shared__ int esh[16], ssh[16], dsh[16];
  __shared__ float srs[32 * 8];
  const int r = trel[blockIdx.x];
  if (r < 0) return;
  const int lane = threadIdx.x;
  if (lane < 16) {
    const int e = eperm[blockIdx.x * 16 + lane];
    esh[lane] = e;
    int s = -1, d = -1;
    if (e >= 0) { s = adj[e]; d = adj[E + e]; }
    ssh[lane] = s; dsh[lane] = d;
  }
  __syncthreads();
#pragma unroll
  for (int m = 0; m < 16; ++m) {
    const int s = ssh[m];
    float4 a = make_float4(0.f, 0.f, 0.f, 0.f);
    if (s >= 0) a = ((const float4*)kb)[(size_t)s * 32 + lane];
    *(float4*)&shk[m][lane * 4] = a;
    const int d = dsh[m];
    float4 b = make_float4(0.f, 0.f, 0.f, 0.f);
    if (d >= 0) b = ((const float4*)qb)[(size_t)d * 32 + lane];
    *(float4*)&shq[m][lane * 4] = b;
  }
  __syncthreads();
  const float* W = ratt + (size_t)r * IN * OUT;
  __builtin_prefetch(W, 0, 3);
  const int row = lane & 15, hi = lane >> 4;
  float rs[8] = {0, 0, 0, 0, 0, 0, 0, 0};
  for (int c = 0; c < 8; ++c) {
    v8f acc = wmma_tile_f32(W, shk, lane, c);     // P = K_src @ A_r (tile)
#pragma unroll
    for (int i = 0; i < 8; ++i) {
      const int m = i + (hi << 3);
      rs[i] += acc[i] * shq[m][c * 16 + row];     // row-dot with q_dst
    }
  }
#pragma unroll
  for (int i = 0; i < 8; ++i) srs[lane * 8 + i] = rs[i];
  __syncthreads();
  if (lane < 16) {
    const int m = lane;
    const int half = (m < 8) ? 0 : 16;
    const int idx = (m < 8) ? m : (m - 8);
    float s = 0.f;
    for (int l = 0; l < 16; ++l) s += srs[(half + l) * 8 + idx];
    const int e = esh[m];
    if (e >= 0) {
      const float scv = s * pri[r] * INV_SQRT_DK;
      sc[e] = scv;
      const int seg = dsh[m] * R + r;
      atomicMax(&mu[seg], fkey(scv));
    }
  }
}

// ---- softmax: exp + segment denominators -----------------------------------
__global__ void hgt_softmax_norm(const int* __restrict__ adj,
                                 const int* __restrict__ edge_type,
                                 const unsigned* __restrict__ mu,
                                 float* __restrict__ sc,
                                 float* __restrict__ den) {
  const int e = blockIdx.x * blockDim.x + threadIdx.x;
  if (e >= E) return;
  const int seg = adj[E + e] * R + edge_type[e];
  const float ex = __expf(sc[e] - fdec(mu[seg]));
  sc[e] = ex;
  atomicAdd(&den[seg], ex);
}

// ---- messages + aggregation: per-relation 16-edge tiles --------------------
__global__ __launch_bounds__(32)
void hgt_msg(const float* __restrict__ vb, const int* __restrict__ adj,
             const int* __restrict__ eperm, const int* __restrict__ trel,
             const float* __restrict__ rmsg, const float* __restrict__ sc,
             const float* __restrict__ den, float* __restrict__ agg) {
  __shared__ float sh[16][132];
  __shared__ int esh[16], ssh[16], dsh[16];
  __shared__ float alsh[16];
  const int r = trel[blockIdx.x];
  if (r < 0) return;
  const int lane = threadIdx.x;
  if (lane < 16) {
    const int e = eperm[blockIdx.x * 16 + lane];
    esh[lane] = e;
    int s = -1, d = -1; float a = 0.f;
    if (e >= 0) {
      s = adj[e]; d = adj[E + e];
      a = sc[e] / den[d * R + r];                 // alpha
    }
    ssh[lane] = s; dsh[lane] = d; alsh[lane] = a;
  }
  __syncthreads();
#pragma unroll
  for (int m = 0; m < 16; ++m) {
    const int s = ssh[m];
    float4 val = make_float4(0.f, 0.f, 0.f, 0.f);
    if (s >= 0) val = ((const float4*)vb)[(size_t)s * 32 + lane];
    *(float4*)&sh[m][lane * 4] = val;
  }
  __syncthreads();
  const float* W = rmsg + (size_t)r * IN * OUT;
  __builtin_prefetch(W, 0, 3);
  const int row = lane & 15, hi = lane >> 4;
  for (int c = 0; c < 8; ++c) {
    v8f acc = wmma_tile_f32(W, sh, lane, c);      // vM = V_src @ M_r (tile)
#pragma unroll
    for (int i = 0; i < 8; ++i) {
      const int m = i + (hi << 3);
      const int d = dsh[m];
      if (d >= 0)
        atomicAdd(&agg[(size_t)d * OUT + c * 16 + row], acc[i] * alsh[m]);
    }
  }
}

// ---- output transform: per-type 16-node tiles, sigmoid(skip) gate ----------
__global__ __launch_bounds__(32)
void hgt_out(const float* __restrict__ agg, const int* __restrict__ nperm,
             const int* __restrict__ ttype, const float* __restrict__ aW,
             const float* __restrict__ skip, float* __restrict__ out) {
  __shared__ float sh[16][132];
  __shared__ int nd[16];
  const int t = ttype[blockIdx.x];
  if (t < 0) return;
  const int lane = threadIdx.x;
  if (lane < 16) nd[lane] = nperm[blockIdx.x * 16 + lane];
  __syncthreads();
#pragma unroll
  for (int m = 0; m < 16; ++m) {
    const int n = nd[m];
    float4 val = make_float4(0.f, 0.f, 0.f, 0.f);
    if (n >= 0) val = ((const float4*)agg)[(size_t)n * 32 + lane];
    *(float4*)&sh[m][lane * 4] = val;
  }
  __syncthreads();
  const float* W = aW + (size_t)t * OUT * OUT;
  __builtin_prefetch(W, 0, 3);
  const float gate = 1.f / (1.f + __expf(-skip[t]));
  const int row = lane & 15, hi = lane >> 4;
  for (int c = 0; c < 8; ++c) {
    v8f acc = wmma_tile_f32(W, sh, lane, c);
#pragma unroll
    for (int i = 0; i < 8; ++i) {
      const int m = i + (hi << 3);
      const int n = nd[m];
      if (n >= 0) out[(size_t)n * OUT + c * 16 + row] = acc[i] * gate;
    }
  }
}

// ---------------------------------------------------------------------------
extern "C" void kernel_launch(void* const* d_in, const int* in_sizes, int n_in,
                              void* d_out, int out_size, void* d_ws,
                              size_t ws_size, hipStream_t stream) {
  (void)in_sizes; (void)n_in; (void)out_size;
  if (ws_size < WS_NEEDED) return;  // defensive: avoid scribbling past scratch

  const float* h         = (const float*)d_in[0];
  const int*   adj       = (const int*)d_in[1];
  const int*   edge_type = (const int*)d_in[2];
  const int*   node_type = (const int*)d_in[3];
  const float* kW        = (const float*)d_in[6];
  const float* qW        = (const float*)d_in[7];
  const float* vW        = (const float*)d_in[8];
  const float* aW        = (const float*)d_in[9];
  const float* pri       = (const float*)d_in[10];
  const float* ratt      = (const float*)d_in[11];
  const float* rmsg      = (const float*)d_in[12];
  const float* skip      = (const float*)d_in[13];
  float* out = (float*)d_out;

  char* ws = (char*)d_ws;
  float*    kb    = (float*)(ws + OFF_K);
  float*    qb    = (float*)(ws + OFF_Q);
  float*    vb    = (float*)(ws + OFF_V);
  float*    agg   = (float*)(ws + OFF_AGG);
  float*    sc    = (float*)(ws + OFF_SC);
  unsigned* mu    = (unsigned*)(ws + OFF_MU);
  float*    den   = (float*)(ws + OFF_DEN);
  int*      cnt   = (int*)(ws + OFF_CNT);
  int*      nperm = (int*)(ws + OFF_NPERM);
  int*      ttype = (int*)(ws + OFF_TTYPE);
  int*      eperm = (int*)(ws + OFF_EPERM);
  int*      trel  = (int*)(ws + OFF_TREL);

  hgt_init<<<1024, 256, 0, stream>>>(agg, mu, den, cnt, nperm, eperm, ttype, trel);
  hgt_count<<<(E + 255) / 256, 256, 0, stream>>>(node_type, edge_type, cnt);
  hgt_scan<<<1, 256, 0, stream>>>(cnt, ttype, trel);
  hgt_scatter<<<(E + 255) / 256, 256, 0, stream>>>(node_type, edge_type, cnt,
                                                   nperm, eperm);
  hgt_kqv<<<NODE_TILES, 32, 0, stream>>>(h, nperm, ttype, kW, qW, vW, kb, qb, vb);
  hgt_score<<<EDGE_TILES, 32, 0, stream>>>(kb, qb, adj, eperm, trel, ratt, pri,
                                           sc, mu);
  hgt_softmax_norm<<<(E + 255) / 256, 256, 0, stream>>>(adj, edge_type, mu, sc,
                                                        den);
  hgt_msg<<<EDGE_TILES, 32, 0, stream>>>(vb, adj, eperm, trel, rmsg, sc, den,
                                         agg);
  hgt_out<<<NODE_TILES, 32, 0, stream>>>(agg, nperm, ttype, aW, skip, out);
}